// GINModel_88424786690458
// MI455X (gfx1250) — compile-verified
//
#include <hip/hip_runtime.h>
#include <hip/hip_bf16.h>

// GIN: 2x (scatter-sum aggregate -> MLP) on 100k nodes / 1.6M edges, f32.
// agg is initialized with h itself so z = agg + h needs no extra pass.
// GEMMs use V_WMMA_F32_16X16X4_F32 (exact f32 path on CDNA5), K/N templated
// so the K-loop fully unrolls into straight-line WMMA code.

#define N_NODES 100000
#define N_EDGES 1600000
#define IN_C    64
#define HID     128
#define OUT_C   64

typedef __attribute__((ext_vector_type(2))) float floatx2;
typedef __attribute__((ext_vector_type(8))) float floatx8;

// ---------------------------------------------------------------- copy (init agg = h)
__global__ void copy_f4_kernel(float* __restrict__ dst, const float* __restrict__ src,
                               int n4) {
    int i = blockIdx.x * blockDim.x + threadIdx.x;
    if (i < n4) {
        ((float4*)dst)[i] = ((const float4*)src)[i];
    }
}

// ---------------------------------------------------------------- scatter-add edges
// one thread = one (edge, 4-feature chunk); float4 gather + 4 f32 atomics.
// C4 is a power of two so tid/C4 is a shift.
template <int C4>
__global__ void scatter_add_kernel(float* __restrict__ agg, const float* __restrict__ h,
                                   const int* __restrict__ srcIdx,
                                   const int* __restrict__ dstIdx,
                                   int total) {
    int tid = blockIdx.x * blockDim.x + threadIdx.x;
    if (tid >= total) return;
    int e  = tid / C4;
    int c  = (tid & (C4 - 1)) * 4;
    int s  = srcIdx[e];
    int d  = dstIdx[e];
    const float4 v = *(const float4*)(h + (size_t)s * (C4 * 4) + c);
    float* p = agg + (size_t)d * (C4 * 4) + c;
    atomicAdd(p + 0, v.x);
    atomicAdd(p + 1, v.y);
    atomicAdd(p + 2, v.z);
    atomicAdd(p + 3, v.w);
}

// ---------------------------------------------------------------- WMMA f32 GEMM
// C[M x N] = act(A[M x K] @ W[K x N] + bias), one wave per 16x16 output tile.
// A tile staged in LDS (row stride 129 floats to avoid bank conflicts).
// K, N, RELU compile-time -> fully unrolled WMMA chain, immediate-offset loads.
template <int K, int N, int RELU>
__global__ void __launch_bounds__(32)
gemm_bias_act_kernel(const float* __restrict__ A, const float* __restrict__ W,
                     const float* __restrict__ bias, float* __restrict__ C) {
    __shared__ float sA[16 * 129];

    const int mt   = blockIdx.x;      // M tile (16 rows)
    const int nt   = blockIdx.y;      // N tile (16 cols)
    const int lane = threadIdx.x;     // 0..31 (wave32)
    const int half = lane >> 4;       // 0 or 1
    const int l16  = lane & 15;

    // cooperative, coalesced load of the 16 x K A-tile into LDS
    const float4* A4 = (const float4*)(A + (size_t)mt * (16 * K));
    constexpr int K4   = K / 4;       // float4s per row (16 or 32)
    constexpr int TOTV = 16 * K4;
#pragma unroll
    for (int i = lane; i < TOTV; i += 32) {
        int r = i / K4;
        int c = (i & (K4 - 1)) * 4;
        float4 v = A4[i];
        float* dsti = &sA[r * 129 + c];
        dsti[0] = v.x; dsti[1] = v.y; dsti[2] = v.z; dsti[3] = v.w;
    }
    __syncthreads();   // single wave: DS ops in-order; barrier is cheap/NOP

    floatx8 acc = {};
    const float* Wbase = W + nt * 16 + l16 + half * 2 * N;
    const float* sArow = &sA[l16 * 129 + half * 2];

#pragma unroll
    for (int k = 0; k < K; k += 4) {
        // A fragment 16x4: lane half selects K pair {0,1} vs {2,3}, row = lane%16
        floatx2 a;
        a.x = sArow[k + 0];
        a.y = sArow[k + 1];
        // B fragment 4x16: VGPR v holds row k+v (halves offset by 2), col = lane%16
        floatx2 b;
        b.x = Wbase[(k + 0) * N];
        b.y = Wbase[(k + 1) * N];
        acc = __builtin_amdgcn_wmma_f32_16x16x4_f32(
            /*neg_a=*/false, a, /*neg_b=*/false, b,
            /*c_mod=*/(short)0, acc, /*reuse_a=*/false, /*reuse_b=*/false);
    }

    // D layout: VGPR v -> row (v + 8*half), col = lane%16
    const float bv = bias[nt * 16 + l16];
    float* Cout = C + ((size_t)mt * 16 + half * 8) * N + nt * 16 + l16;
#pragma unroll
    for (int v = 0; v < 8; ++v) {
        float x = acc[v] + bv;
        if (RELU) x = fmaxf(x, 0.0f);
        Cout[(size_t)v * N] = x;
    }
}

// ---------------------------------------------------------------- launch
extern "C" void kernel_launch(void* const* d_in, const int* in_sizes, int n_in,
                              void* d_out, int out_size, void* d_ws, size_t ws_size,
                              hipStream_t stream) {
    (void)in_sizes; (void)n_in; (void)out_size; (void)ws_size;

    const float* x    = (const float*)d_in[0];
    const int*   ei   = (const int*)d_in[1];
    const float* W1a  = (const float*)d_in[2];
    const float* b1a  = (const float*)d_in[3];
    const float* W1b  = (const float*)d_in[4];
    const float* b1b  = (const float*)d_in[5];
    const float* W2a  = (const float*)d_in[6];
    const float* b2a  = (const float*)d_in[7];
    const float* W2b  = (const float*)d_in[8];
    const float* b2b  = (const float*)d_in[9];
    float* out = (float*)d_out;

    const int* srcIdx = ei;             // edge_index[0]
    const int* dstIdx = ei + N_EDGES;   // edge_index[1]

    // workspace: three rotating node-feature buffers (128-wide each)
    const size_t buf_elems = (size_t)N_NODES * HID;
    float* B0 = (float*)d_ws;
    float* B1 = B0 + buf_elems;
    float* B2 = B1 + buf_elems;

    const int TPB = 256;
    const int MT  = N_NODES / 16;       // 6250, exact

    // ---- layer 1: agg1 = x ; agg1 += x[src] scattered to dst
    {
        int n4 = N_NODES * IN_C / 4;
        copy_f4_kernel<<<(n4 + TPB - 1) / TPB, TPB, 0, stream>>>(B0, x, n4);
        int total = N_EDGES * (IN_C / 4);
        scatter_add_kernel<IN_C / 4><<<(total + TPB - 1) / TPB, TPB, 0, stream>>>(
            B0, x, srcIdx, dstIdx, total);
    }
    // t1 = relu(agg1 @ W1a + b1a)          [100000x64 @ 64x128]
    gemm_bias_act_kernel<IN_C, HID, 1>
        <<<dim3(MT, HID / 16), 32, 0, stream>>>(B0, W1a, b1a, B1);
    // h1 = relu(t1 @ W1b + b1b)            [100000x128 @ 128x128]
    gemm_bias_act_kernel<HID, HID, 1>
        <<<dim3(MT, HID / 16), 32, 0, stream>>>(B1, W1b, b1b, B2);

    // ---- layer 2: agg2 = h1 ; agg2 += h1[src]
    {
        int n4 = N_NODES * HID / 4;
        copy_f4_kernel<<<(n4 + TPB - 1) / TPB, TPB, 0, stream>>>(B0, B2, n4);
        int total = N_EDGES * (HID / 4);
        scatter_add_kernel<HID / 4><<<(total + TPB - 1) / TPB, TPB, 0, stream>>>(
            B0, B2, srcIdx, dstIdx, total);
    }
    // t2 = relu(agg2 @ W2a + b2a)          [100000x128 @ 128x128]
    gemm_bias_act_kernel<HID, HID, 1>
        <<<dim3(MT, HID / 16), 32, 0, stream>>>(B0, W2a, b2a, B1);
    // out = t2 @ W2b + b2b                 [100000x128 @ 128x64]
    gemm_bias_act_kernel<HID, OUT_C, 0>
        <<<dim3(MT, OUT_C / 16), 32, 0, stream>>>(B1, W2b, b2b, out);
}